// VMFactorizedVoxelGrid_48223892799832
// MI455X (gfx1250) — compile-verified
//
#include <hip/hip_runtime.h>

// ---------------------------------------------------------------------------
// VM-factorized voxel grid (TensoRF-style) for gfx1250 / MI455X.
//   G=1, N=2097152, C=16, R=320, NCOMP=48, F=27
// Strategy:
//   1) transpose planes (C,R,R)->(R,R,C) and lines (C,R)->(R,C) into d_ws so
//      every bilinear corner fetch is 64 contiguous bytes (planes fit in L2).
//   2) per wave: 32 points -> feats[32][48] in LDS -> 48x27 fp32 matmul via
//      V_WMMA_F32_16X16X4_F32 (2 M-tiles x 2 N-tiles x 12 K-chunks).
// ---------------------------------------------------------------------------

typedef float v2f __attribute__((ext_vector_type(2)));
typedef float v8f __attribute__((ext_vector_type(8)));

#define RR 320
#define PPOS (RR * RR)                 // 102400 positions per plane
#define PLANE_T_FLOATS (PPOS * 16)     // 1,638,400 floats per transposed plane
#define LINE_T_FLOATS (RR * 16)        // 5,120 floats per transposed line
#define FSTRIDE 52                     // feats LDS stride (floats); 52*4B % 16 == 0, conflict-friendly

// ---------------------------------------------------------------------------
// Kernel 1: layout transform into workspace.
// ws layout (floats): [planeXY][planeYZ][planeXZ][lineX][lineY][lineZ]
// ---------------------------------------------------------------------------
__global__ __launch_bounds__(256) void vm_transpose_kernel(
    const float* __restrict__ pXY, const float* __restrict__ pYZ,
    const float* __restrict__ pXZ, const float* __restrict__ vX,
    const float* __restrict__ vY, const float* __restrict__ vZ,
    float* __restrict__ ws) {
  int tid = blockIdx.x * 256 + threadIdx.x;
  if (tid < 3 * PPOS) {
    int pl = tid / PPOS;
    int pos = tid - pl * PPOS;
    const float* src = (pl == 0) ? pXY : (pl == 1) ? pYZ : pXZ;
    float* dst = ws + (size_t)pl * PLANE_T_FLOATS + (size_t)pos * 16;
#pragma unroll
    for (int c = 0; c < 16; ++c) dst[c] = src[(size_t)c * PPOS + pos];
  } else {
    int idx = tid - 3 * PPOS;
    if (idx < 3 * RR) {
      int ln = idx / RR;
      int x = idx - ln * RR;
      const float* src = (ln == 0) ? vX : (ln == 1) ? vY : vZ;
      float* dst = ws + (size_t)3 * PLANE_T_FLOATS + (size_t)ln * LINE_T_FLOATS + x * 16;
#pragma unroll
      for (int c = 0; c < 16; ++c) dst[c] = src[c * RR + x];
    }
  }
}

// ---------------------------------------------------------------------------
// Interpolation helpers (align_corners=True, zero padding, fp32).
// ---------------------------------------------------------------------------
__device__ __forceinline__ float4 mul4(float4 a, float4 b) {
  float4 o;
  o.x = a.x * b.x; o.y = a.y * b.y; o.z = a.z * b.z; o.w = a.w * b.w;
  return o;
}

__device__ __forceinline__ void sample_plane(const float* __restrict__ pt,
                                             float u, float v, float4 out[4]) {
  // u indexes W (x), v indexes H (y); storage (y*RR + x)*16 + c
  float x = (u + 1.0f) * (0.5f * (RR - 1));
  float y = (v + 1.0f) * (0.5f * (RR - 1));
  float x0f = floorf(x), y0f = floorf(y);
  float wx = x - x0f, wy = y - y0f;
  int x0 = (int)x0f, y0 = (int)y0f;
  int x1 = x0 + 1, y1 = y0 + 1;
  float mx0 = (x0 >= 0 && x0 < RR) ? 1.0f : 0.0f;
  float mx1 = (x1 >= 0 && x1 < RR) ? 1.0f : 0.0f;
  float my0 = (y0 >= 0 && y0 < RR) ? 1.0f : 0.0f;
  float my1 = (y1 >= 0 && y1 < RR) ? 1.0f : 0.0f;
  int x0c = min(max(x0, 0), RR - 1), x1c = min(max(x1, 0), RR - 1);
  int y0c = min(max(y0, 0), RR - 1), y1c = min(max(y1, 0), RR - 1);
  float w00 = (1.0f - wx) * (1.0f - wy) * mx0 * my0;
  float w10 = wx * (1.0f - wy) * mx1 * my0;
  float w01 = (1.0f - wx) * wy * mx0 * my1;
  float w11 = wx * wy * mx1 * my1;
  const float4* c00 = (const float4*)(pt + (y0c * RR + x0c) * 16);
  const float4* c10 = (const float4*)(pt + (y0c * RR + x1c) * 16);
  const float4* c01 = (const float4*)(pt + (y1c * RR + x0c) * 16);
  const float4* c11 = (const float4*)(pt + (y1c * RR + x1c) * 16);
#pragma unroll
  for (int cg = 0; cg < 4; ++cg) {
    float4 a = c00[cg], b = c10[cg], c = c01[cg], d = c11[cg];
    float4 o;
    o.x = fmaf(w11, d.x, fmaf(w01, c.x, fmaf(w10, b.x, w00 * a.x)));
    o.y = fmaf(w11, d.y, fmaf(w01, c.y, fmaf(w10, b.y, w00 * a.y)));
    o.z = fmaf(w11, d.z, fmaf(w01, c.z, fmaf(w10, b.z, w00 * a.z)));
    o.w = fmaf(w11, d.w, fmaf(w01, c.w, fmaf(w10, b.w, w00 * a.w)));
    out[cg] = o;
  }
}

__device__ __forceinline__ void sample_line(const float* __restrict__ ln,
                                            float t, float4 out[4]) {
  float x = (t + 1.0f) * (0.5f * (RR - 1));
  float x0f = floorf(x);
  float wx = x - x0f;
  int x0 = (int)x0f, x1 = x0 + 1;
  float m0 = (x0 >= 0 && x0 < RR) ? 1.0f : 0.0f;
  float m1 = (x1 >= 0 && x1 < RR) ? 1.0f : 0.0f;
  int x0c = min(max(x0, 0), RR - 1), x1c = min(max(x1, 0), RR - 1);
  float w0 = (1.0f - wx) * m0;
  float w1 = wx * m1;
  const float4* t0 = (const float4*)(ln + x0c * 16);
  const float4* t1 = (const float4*)(ln + x1c * 16);
#pragma unroll
  for (int cg = 0; cg < 4; ++cg) {
    float4 a = t0[cg], b = t1[cg];
    float4 o;
    o.x = fmaf(w1, b.x, w0 * a.x);
    o.y = fmaf(w1, b.y, w0 * a.y);
    o.z = fmaf(w1, b.z, w0 * a.z);
    o.w = fmaf(w1, b.w, w0 * a.w);
    out[cg] = o;
  }
}

// ---------------------------------------------------------------------------
// Kernel 2: main. 256 threads = 8 waves. Each wave handles 32 points.
// feats[32][48] -> LDS -> WMMA f32 16x16x4 matmul with basis (48x27, N padded
// to 32). EXEC stays all-ones (grid exactly covers N; no divergence before the
// WMMA loop).
// ---------------------------------------------------------------------------
__global__ __launch_bounds__(256) void vm_main_kernel(
    const float* __restrict__ points, const float* __restrict__ basis,
    const float* __restrict__ ws, float* __restrict__ out) {
  __shared__ __align__(16) float ldsF[8 * 32 * FSTRIDE];  // 53,248 B
  __shared__ __align__(16) float ldsB[32 * 48];           //  6,144 B

  const float* wsXY = ws;
  const float* wsYZ = ws + (size_t)PLANE_T_FLOATS;
  const float* wsXZ = ws + (size_t)2 * PLANE_T_FLOATS;
  const float* wsLX = ws + (size_t)3 * PLANE_T_FLOATS;
  const float* wsLY = wsLX + LINE_T_FLOATS;
  const float* wsLZ = wsLY + LINE_T_FLOATS;

  int tid = threadIdx.x;
  int lane = tid & 31;
  int wave = tid >> 5;

  // Stage basis as B^T-ish: ldsB[n*48 + k] = basis[k*27 + n], zero-pad n>=27.
  for (int i = tid; i < 32 * 48; i += 256) {
    int n = i / 48;
    int k = i - n * 48;
    ldsB[i] = (n < 27) ? basis[k * 27 + n] : 0.0f;
  }

  size_t gp = (size_t)blockIdx.x * 256 + tid;  // one point per lane
  float px = points[gp * 3 + 0];
  float py = points[gp * 3 + 1];
  float pz = points[gp * 3 + 2];

  float4 Pxy[4], Pxz[4], Pyz[4], Lx[4], Ly[4], Lz[4];
  sample_plane(wsXY, px, py, Pxy);   // grid x = px, grid y = py
  sample_plane(wsXZ, px, pz, Pxz);   // grid x = px, grid y = pz
  sample_plane(wsYZ, py, pz, Pyz);   // grid x = py, grid y = pz
  sample_line(wsLX, px, Lx);
  sample_line(wsLY, py, Ly);
  sample_line(wsLZ, pz, Lz);

  float* fr = &ldsF[wave * (32 * FSTRIDE) + lane * FSTRIDE];
#pragma unroll
  for (int cg = 0; cg < 4; ++cg) {
    *(float4*)&fr[cg * 4]      = mul4(Pxy[cg], Lz[cg]);  // feats[ 0..15]
    *(float4*)&fr[16 + cg * 4] = mul4(Pxz[cg], Ly[cg]);  // feats[16..31]
    *(float4*)&fr[32 + cg * 4] = mul4(Pyz[cg], Lx[cg]);  // feats[32..47]
  }
  __syncthreads();

  // WMMA: D(16x16) = A(16x4) x B(4x16) + C, fp32. 12 K-chunks over K=48.
  v8f d00 = {}, d01 = {}, d10 = {}, d11 = {};
  const float* fw = &ldsF[wave * (32 * FSTRIDE)];
  int n0 = lane & 15;                       // N column for this lane
  int kofs = (lane & 16) ? 2 : 0;           // lanes 16-31 carry K=2,3 of chunk
  int prow0 = (lane & 15) * FSTRIDE;        // M-tile 0: points 0..15
  int prow1 = prow0 + 16 * FSTRIDE;         // M-tile 1: points 16..31
#pragma unroll
  for (int kk = 0; kk < 12; ++kk) {
    int k = kk * 4 + kofs;
    v2f a0 = *(const v2f*)&fw[prow0 + k];
    v2f a1 = *(const v2f*)&fw[prow1 + k];
    v2f b0 = *(const v2f*)&ldsB[n0 * 48 + k];
    v2f b1 = *(const v2f*)&ldsB[(n0 + 16) * 48 + k];
    d00 = __builtin_amdgcn_wmma_f32_16x16x4_f32(false, a0, false, b0, (short)0, d00, false, false);
    d01 = __builtin_amdgcn_wmma_f32_16x16x4_f32(false, a0, false, b1, (short)0, d01, false, false);
    d10 = __builtin_amdgcn_wmma_f32_16x16x4_f32(false, a1, false, b0, (short)0, d10, false, false);
    d11 = __builtin_amdgcn_wmma_f32_16x16x4_f32(false, a1, false, b1, (short)0, d11, false, false);
  }

  // D layout: VGPR v -> M = v (lanes 0-15) or 8+v (lanes 16-31), N = lane%16.
  size_t base = (size_t)blockIdx.x * 256 + (size_t)wave * 32;
  int mofs = (lane & 16) ? 8 : 0;
  int f1 = n0 + 16;
#pragma unroll
  for (int v = 0; v < 8; ++v) {
    size_t pA = base + mofs + v;        // M-tile 0 point
    size_t pB = base + 16 + mofs + v;   // M-tile 1 point
    out[pA * 27 + n0] = d00[v];
    out[pB * 27 + n0] = d10[v];
    if (f1 < 27) {
      out[pA * 27 + f1] = d01[v];
      out[pB * 27 + f1] = d11[v];
    }
  }
}

// ---------------------------------------------------------------------------
// Launch. Input order per setup_inputs():
//   0 points (G,N,3)  1 xy-plane  2 yz-plane  3 xz-plane
//   4 x-line  5 y-line  6 z-line  7 basis (48,27)
// ws usage: 3*1,638,400 + 3*5,120 floats = 19,722,240 bytes.
// ---------------------------------------------------------------------------
extern "C" void kernel_launch(void* const* d_in, const int* in_sizes, int n_in,
                              void* d_out, int out_size, void* d_ws, size_t ws_size,
                              hipStream_t stream) {
  const float* points = (const float*)d_in[0];
  const float* pXY = (const float*)d_in[1];
  const float* pYZ = (const float*)d_in[2];
  const float* pXZ = (const float*)d_in[3];
  const float* vX = (const float*)d_in[4];
  const float* vY = (const float*)d_in[5];
  const float* vZ = (const float*)d_in[6];
  const float* basis = (const float*)d_in[7];
  float* ws = (float*)d_ws;
  float* out = (float*)d_out;

  int total_t = 3 * PPOS + 3 * RR;  // 308,160 transform threads
  int tb = (total_t + 255) / 256;
  vm_transpose_kernel<<<tb, 256, 0, stream>>>(pXY, pYZ, pXZ, vX, vY, vZ, ws);

  int N = in_sizes[0] / 3;          // 2,097,152 (divisible by 256)
  int blocks = N / 256;
  vm_main_kernel<<<blocks, 256, 0, stream>>>(points, basis, ws, out);
}